// graph_convolutional_layer_1915555414709
// MI455X (gfx1250) — compile-verified
//
#include <hip/hip_runtime.h>

typedef float v2f __attribute__((ext_vector_type(2)));
typedef float v8f __attribute__((ext_vector_type(8)));

#define D_IN  256
#define D_OUT 64

// ---------------------------------------------------------------------------
// out[i*64 + j] = b[j]   (bias broadcast; atomics accumulate on top of this)
// ---------------------------------------------------------------------------
__global__ __launch_bounds__(256) void gcn_init_bias(float* __restrict__ out,
                                                     const float* __restrict__ b,
                                                     int total) {
  int i = blockIdx.x * blockDim.x + threadIdx.x;
  if (i < total) out[i] = b[i & (D_OUT - 1)];
}

// ---------------------------------------------------------------------------
// XW = X @ W  via V_WMMA_F32_16X16X4_F32 (fp32 WMMA -> full precision).
// One wave32 computes a 16x64 stripe: 4 accumulators (v8f), K-loop step 4.
//
// W is staged in LDS in K-PAIR-INTERLEAVED layout:
//   Wlds[p*64 + n] = { W[2p][n], W[2p+1][n] }   (float2, p = 0..127)
// so every WMMA B fragment {W[k][n], W[k+1][n]} is a single aligned
// ds_load_b64 straight into an even VGPR pair -> no repacking movs.
//
// A (16x4 f32) per-lane layout (ISA 7.12.2): lane m (0-15) VGPR{0,1} = K{0,1},
// lane m+16 VGPR{0,1} = K{2,3}.  C/D: VGPR j -> row j (lanes 0-15) /
// row j+8 (lanes 16-31), col = lane&15.
// ---------------------------------------------------------------------------
__global__ __launch_bounds__(256) void gcn_gemm_wmma(const float* __restrict__ X,
                                                     const float* __restrict__ W,
                                                     float* __restrict__ XW,
                                                     int n_nodes) {
  __shared__ float2 Wlds[(D_IN / 2) * D_OUT];  // 64 KB of the 320 KB/WGP

  // Cooperative copy of W into LDS, interleaving K pairs (n fastest -> coalesced).
  for (int idx = threadIdx.x; idx < (D_IN / 2) * D_OUT; idx += blockDim.x) {
    int p = idx >> 6;          // K-pair index
    int n = idx & 63;          // column
    float2 w2;
    w2.x = W[(size_t)(2 * p)     * D_OUT + n];
    w2.y = W[(size_t)(2 * p + 1) * D_OUT + n];
    Wlds[idx] = w2;
  }
  __syncthreads();

  const int lane = threadIdx.x & 31;
  const int wave = threadIdx.x >> 5;          // 8 waves / block
  const int row0 = (blockIdx.x * 8 + wave) * 16;
  if (row0 >= n_nodes) return;                // wave-uniform: EXEC stays all-1s

  const int m  = lane & 15;                   // A: M row; B/D: N column
  const int kh = (lane >> 4) * 2;             // per-half K offset: 0 or 2

  v8f acc0 = {}, acc1 = {}, acc2 = {}, acc3 = {};
  const float* xrow = X + (size_t)(row0 + m) * D_IN;

  #pragma unroll 4
  for (int kb = 0; kb < D_IN; kb += 4) {
    // A fragment: two consecutive K values for this lane's row (8B load)
    v2f a = *(const v2f*)(xrow + kb + kh);

    // B fragments for the 4 N-tiles: one ds_load_b64 each, pre-paired in LDS
    const float2* wb = Wlds + ((kb + kh) >> 1) * D_OUT + m;
    v2f b0 = *(const v2f*)(wb + 0);
    v2f b1 = *(const v2f*)(wb + 16);
    v2f b2 = *(const v2f*)(wb + 32);
    v2f b3 = *(const v2f*)(wb + 48);

    acc0 = __builtin_amdgcn_wmma_f32_16x16x4_f32(false, a, false, b0, (short)0, acc0, false, false);
    acc1 = __builtin_amdgcn_wmma_f32_16x16x4_f32(false, a, false, b1, (short)0, acc1, false, false);
    acc2 = __builtin_amdgcn_wmma_f32_16x16x4_f32(false, a, false, b2, (short)0, acc2, false, false);
    acc3 = __builtin_amdgcn_wmma_f32_16x16x4_f32(false, a, false, b3, (short)0, acc3, false, false);
  }

  // Store: VGPR j -> row (row0 + j + 8*(lane>=16)), col = tile*16 + (lane&15).
  // b32 stores; 16 consecutive lanes cover a contiguous 64B run.
  float* ob = XW + (size_t)(row0 + (lane >> 4) * 8) * D_OUT + m;
  #pragma unroll
  for (int j = 0; j < 8; ++j) {
    ob[(size_t)j * D_OUT + 0]  = acc0[j];
    ob[(size_t)j * D_OUT + 16] = acc1[j];
    ob[(size_t)j * D_OUT + 32] = acc2[j];
    ob[(size_t)j * D_OUT + 48] = acc3[j];
  }
}

// ---------------------------------------------------------------------------
// COO scatter: out[row[e]][:] += vals[e] * XW[col[e]][:]
// One wave32 per edge; each lane owns a float2 (64 cols / 32 lanes).
// e is wave-uniform -> readfirstlane makes row/col/vals scalar (s_load, KMcnt),
// leaving the vector-memory path for the 256B gather + 2 atomics per lane.
// XW (12.8 MB) and out (12.8 MB) are both resident in the 192 MB L2, so the
// ~410 MB of gather+atomic traffic never leaves L2. atomicAdd with unused
// return lowers to no-return global_atomic_add_f32 (STOREcnt path).
// ---------------------------------------------------------------------------
__global__ __launch_bounds__(256) void gcn_spmm_scatter(const float* __restrict__ XW,
                                                        const int* __restrict__ row,
                                                        const int* __restrict__ col,
                                                        const float* __restrict__ vals,
                                                        float* __restrict__ out,
                                                        int n_edges) {
  int gid  = blockIdx.x * blockDim.x + threadIdx.x;
  int e    = __builtin_amdgcn_readfirstlane(gid >> 5);  // wave-uniform edge id
  int lane = threadIdx.x & 31;
  if (e >= n_edges) return;                             // uniform branch

  int   r = row[e];     // scalar loads (uniform address)
  int   c = col[e];
  float v = vals[e];

  const float2* src = (const float2*)(XW + (size_t)c * D_OUT) + lane;  // coalesced 256B/wave
  float2 mv = *src;

  float* dst = out + (size_t)r * D_OUT + lane * 2;
  atomicAdd(dst + 0, v * mv.x);
  atomicAdd(dst + 1, v * mv.y);
}

// ---------------------------------------------------------------------------
extern "C" void kernel_launch(void* const* d_in, const int* in_sizes, int n_in,
                              void* d_out, int out_size, void* d_ws, size_t ws_size,
                              hipStream_t stream) {
  const float* X    = (const float*)d_in[0];
  const int*   row  = (const int*)  d_in[1];
  const int*   col  = (const int*)  d_in[2];
  const float* vals = (const float*)d_in[3];
  const float* W    = (const float*)d_in[4];
  const float* b    = (const float*)d_in[5];
  float* out = (float*)d_out;
  float* XW  = (float*)d_ws;            // needs n_nodes*64*4 = 12.8 MB scratch

  const int n_nodes = in_sizes[0] / D_IN;   // 50000
  const int n_edges = in_sizes[1];          // 800000

  // 1) out = bias (re-initialized every call -> deterministic under replay)
  const int total = n_nodes * D_OUT;
  gcn_init_bias<<<(total + 255) / 256, 256, 0, stream>>>(out, b, total);

  // 2) XW = X @ W  (8 row-tiles of 16 per block)
  const int rowTiles   = (n_nodes + 15) / 16;     // 3125
  const int gemmBlocks = (rowTiles + 7) / 8;      // 391
  gcn_gemm_wmma<<<gemmBlocks, 256, 0, stream>>>(X, W, XW, n_nodes);

  // 3) scatter-accumulate edges (one wave per edge)
  const long long spmmThreads = (long long)n_edges * 32;
  const int spmmBlocks = (int)((spmmThreads + 255) / 256);  // 100000
  gcn_spmm_scatter<<<spmmBlocks, 256, 0, stream>>>(XW, row, col, vals, out, n_edges);
}